// L1Aggregator_85306640433215
// MI455X (gfx1250) — compile-verified
//
#include <hip/hip_runtime.h>

#define N_ROWS   500000
#define DIM      512
#define NTILES   (N_ROWS / 16)      // 31250 exact 16-row tiles
#define GRID_A   1024
#define TSTRIDE  516                // padded LDS row stride (floats) -> conflict-free WMMA B reads
#define TILE_F   (16 * TSTRIDE)     // floats per tile buffer
#define PSTRIDE  520                // floats per block partial: [m, s, r[512], pad]

// dynamic LDS partition (floats): tile0 | tile1 | q[512] | ekred[128] | wexp[16] | scale/pad[8]
#define SMEM_FLOATS (2 * TILE_F + 512 + 128 + 16 + 8)

typedef __attribute__((ext_vector_type(2))) float f32x2;
typedef __attribute__((ext_vector_type(4))) float f32x4;
typedef __attribute__((ext_vector_type(8))) float f32x8;
typedef int v4i __attribute__((vector_size(16)));   // matches builtin's expected pointee

#if defined(__HIP_DEVICE_COMPILE__) && __has_builtin(__builtin_amdgcn_global_load_async_to_lds_b128)
#define HAS_ASYNC 1
#define ASYNC_B128(g, l)                                                     \
    __builtin_amdgcn_global_load_async_to_lds_b128(                          \
        (__attribute__((address_space(1))) v4i*)(g),                         \
        (__attribute__((address_space(3))) v4i*)(l), 0, 0)
#else
#define HAS_ASYNC 0
#endif

#if defined(__HIP_DEVICE_COMPILE__) && __has_builtin(__builtin_amdgcn_s_wait_asynccnt)
#define WAIT_ASYNC(n) __builtin_amdgcn_s_wait_asynccnt(n)
#else
#define WAIT_ASYNC(n) asm volatile("s_wait_asynccnt %0" ::"i"(n) : "memory")
#endif

// ---------------------------------------------------------------------------
// Kernel A: fused online-softmax attention pooling over 16-row tiles.
//   stage : GLOBAL_LOAD_ASYNC_TO_LDS_B128 double-buffered prefetch (ASYNCcnt)
//   pass1 : ek[0..15] = q . tile^T   via v_wmma_f32_16x16x4_f32 (A = q replicated)
//   pass2 : r += exp(ek - m) * tile  from the same LDS tile (fvs read ONCE)
// ---------------------------------------------------------------------------
__global__ __launch_bounds__(256) void attn_pool_partial(
    const float* __restrict__ fvs, const float* __restrict__ q0,
    float* __restrict__ part)
{
    extern __shared__ float smem[];
    float* tbuf  = smem;                    // 2 tile buffers
    float* qsh   = smem + 2 * TILE_F;       // 512
    float* ekred = qsh + 512;               // 128 (8 waves x 16)
    float* wexp  = ekred + 128;             // 16
    float* scsh  = wexp + 16;               // 1

    const int t    = threadIdx.x;
    const int lane = t & 31;
    const int wave = t >> 5;

    qsh[t]       = q0[t];
    qsh[t + 256] = q0[t + 256];
    __syncthreads();

    f32x2 racc = {0.f, 0.f};                // this thread owns cols 2t, 2t+1
    float m_run = -3.0e38f;                 // online-softmax state (wave 0 uniform)
    float s_run = 0.f;

#if HAS_ASYNC
    // ---- prologue: async-prefetch first tile into buffer 0 ----
    int p = 0;
    if (blockIdx.x < NTILES) {
        const float* src = fvs + (long)blockIdx.x * 16 * DIM;
        #pragma unroll
        for (int i = 0; i < 8; ++i) {
            int idx = i * 256 + t, row = idx >> 7, col4 = idx & 127;
            ASYNC_B128(src + row * DIM + col4 * 4, tbuf + row * TSTRIDE + col4 * 4);
        }
    }
#endif

    for (int tileIdx = blockIdx.x; tileIdx < NTILES; tileIdx += gridDim.x) {
#if HAS_ASYNC
        float* tile = tbuf + p * TILE_F;
        // ---- issue async prefetch of next tile into the other buffer ----
        const int nxt = tileIdx + gridDim.x;
        if (nxt < NTILES) {
            const float* nsrc = fvs + (long)nxt * 16 * DIM;
            float* ntile = tbuf + (1 - p) * TILE_F;
            #pragma unroll
            for (int i = 0; i < 8; ++i) {
                int idx = i * 256 + t, row = idx >> 7, col4 = idx & 127;
                ASYNC_B128(nsrc + row * DIM + col4 * 4, ntile + row * TSTRIDE + col4 * 4);
            }
            WAIT_ASYNC(8);   // older batch (this tile) done; next batch stays in flight
        } else {
            WAIT_ASYNC(0);
        }
        __syncthreads();     // all waves' async writes for this tile visible
        p ^= 1;
#else
        float* tile = tbuf;
        // ---- synchronous staging fallback: coalesced b128 via VGPRs ----
        const float* src = fvs + (long)tileIdx * 16 * DIM;
        #pragma unroll
        for (int i = 0; i < 8; ++i) {
            int idx = i * 256 + t, row = idx >> 7, col4 = idx & 127;
            f32x4 v = *(const f32x4*)(src + row * DIM + col4 * 4);
            *(f32x4*)(tile + row * TSTRIDE + col4 * 4) = v;
        }
        __syncthreads();
#endif

        // ---- pass 1: WMMA scores; wave w covers K-chunks [16w, 16w+16) ----
        {
            f32x8 c = {};
            const int half = (lane >= 16) ? 2 : 0;   // K-half split per 32-bit A/B layout
            const int n    = lane & 15;
            #pragma unroll
            for (int kk = 0; kk < 16; ++kk) {
                int kc = wave * 16 + kk;
                // A[m][k] = q[4kc+k] (replicated rows): VGPR0=K0/K2, VGPR1=K1/K3
                f32x2 a = *(const f32x2*)(qsh + kc * 4 + half);
                // B[k][n] = tile[n][4kc+k]: lane n holds column n, K split by lane half
                f32x2 b = *(const f32x2*)(tile + n * TSTRIDE + kc * 4 + half);
                c = __builtin_amdgcn_wmma_f32_16x16x4_f32(
                        false, a, false, b, (short)0, c, false, false);
            }
            // D[m][n] identical over m -> lane n holds partial ek[n] in every C reg
            if (lane < 16) ekred[wave * 16 + lane] = c[0];
        }
        __syncthreads();

        // ---- wave 0: reduce partial scores, online softmax update ----
        if (wave == 0) {
            const int n = lane & 15;
            float ek = 0.f;
            #pragma unroll
            for (int w = 0; w < 8; ++w) ek += ekred[w * 16 + n];
            float mt = ek;
            mt = fmaxf(mt, __shfl_xor(mt, 1, 32));
            mt = fmaxf(mt, __shfl_xor(mt, 2, 32));
            mt = fmaxf(mt, __shfl_xor(mt, 4, 32));
            mt = fmaxf(mt, __shfl_xor(mt, 8, 32));
            float m_new = fmaxf(m_run, mt);
            float sc    = __expf(m_run - m_new);
            float we    = __expf(ek - m_new);
            float sum = we;
            sum += __shfl_xor(sum, 1, 32);
            sum += __shfl_xor(sum, 2, 32);
            sum += __shfl_xor(sum, 4, 32);
            sum += __shfl_xor(sum, 8, 32);
            s_run = s_run * sc + sum;
            m_run = m_new;
            if (lane < 16) wexp[lane] = we;
            if (lane == 0) scsh[0] = sc;
        }
        __syncthreads();

        // ---- pass 2: weighted accumulate from the LDS tile ----
        {
            float sc = scsh[0];
            racc.x *= sc; racc.y *= sc;
            #pragma unroll
            for (int n = 0; n < 16; ++n) {
                float w = wexp[n];
                f32x2 v = *(const f32x2*)(tile + n * TSTRIDE + 2 * t);
                racc.x += w * v.x;
                racc.y += w * v.y;
            }
        }
        __syncthreads();   // buffer reused (async-written) next iteration
    }

    // ---- emit per-block partial {m, s, r[512]} ----
    float* pp = part + (long)blockIdx.x * PSTRIDE;
    if (t == 0) { pp[0] = m_run; pp[1] = s_run; }
    *(f32x2*)(pp + 2 + 2 * t) = racc;
}

// ---------------------------------------------------------------------------
// Kernel B: merge block partials with max renormalization, divide by sum.
// ---------------------------------------------------------------------------
__global__ __launch_bounds__(256) void attn_pool_reduce(
    const float* __restrict__ part, float* __restrict__ out, int nblocks)
{
    __shared__ float red[256];
    __shared__ float wgt[GRID_A];
    __shared__ float Msh, Ssh;

    const int t = threadIdx.x;

    float lm = -3.0e38f;
    for (int b = t; b < nblocks; b += 256)
        lm = fmaxf(lm, part[(long)b * PSTRIDE]);
    red[t] = lm; __syncthreads();
    for (int s = 128; s > 0; s >>= 1) {
        if (t < s) red[t] = fmaxf(red[t], red[t + s]);
        __syncthreads();
    }
    if (t == 0) Msh = red[0];
    __syncthreads();
    const float M = Msh;

    float ls = 0.f;
    for (int b = t; b < nblocks; b += 256) {
        float w = __expf(part[(long)b * PSTRIDE] - M);
        wgt[b] = w;
        ls += w * part[(long)b * PSTRIDE + 1];
    }
    red[t] = ls; __syncthreads();
    for (int s = 128; s > 0; s >>= 1) {
        if (t < s) red[t] += red[t + s];
        __syncthreads();
    }
    if (t == 0) Ssh = red[0];
    __syncthreads();
    const float invS = 1.f / Ssh;

    f32x2 acc = {0.f, 0.f};
    for (int b = 0; b < nblocks; ++b) {
        float w = wgt[b];
        f32x2 v = *(const f32x2*)(part + (long)b * PSTRIDE + 2 + 2 * t);
        acc.x += w * v.x;
        acc.y += w * v.y;
    }
    out[2 * t]     = acc.x * invS;
    out[2 * t + 1] = acc.y * invS;
}

extern "C" void kernel_launch(void* const* d_in, const int* in_sizes, int n_in,
                              void* d_out, int out_size, void* d_ws, size_t ws_size,
                              hipStream_t stream) {
    const float* fvs = (const float*)d_in[0];   // (500000, 512) f32
    const float* q0  = (const float*)d_in[1];   // (1, 512) f32
    float* out  = (float*)d_out;                // (1, 512) f32
    float* part = (float*)d_ws;

    int G = GRID_A;
    int maxG = (int)(ws_size / ((size_t)PSTRIDE * sizeof(float)));
    if (maxG < 1) maxG = 1;
    if (G > maxG) G = maxG;

    const size_t smem_bytes = (size_t)SMEM_FLOATS * sizeof(float);  // ~68.8 KB < 320 KB WGP LDS
    attn_pool_partial<<<G, 256, smem_bytes, stream>>>(fvs, q0, part);
    attn_pool_reduce<<<1, 256, 0, stream>>>(part, out, G);
}